// ISL_44573170598559
// MI455X (gfx1250) — compile-verified
//
#include <hip/hip_runtime.h>

typedef __attribute__((ext_vector_type(2))) float v2f;
typedef __attribute__((ext_vector_type(8))) float v8f;

#define NPTS  4096
#define BATCH 16
#define KNN   20
#define COLS  (BATCH * NPTS)   // 65536
#define CH    64
#define BN_EPS 1e-5f

// ---------------------------------------------------------------------------
// Pass 1: reduce 3-vector first/second moments of x and of edge features
// f = x[:, idx] - x[:, n].  18 fp64 accumulators, deterministic tree reduce.
// grid = 256 blocks x 256 threads  (one thread per (b,n) column)
// ---------------------------------------------------------------------------
__global__ void k_stats_in(const float* __restrict__ x, const int* __restrict__ idx,
                           double* __restrict__ part) {
    int tid = threadIdx.x;
    int col = blockIdx.x * 256 + tid;
    int b = col >> 12, n = col & 4095;
    const float* xb = x + (size_t)b * 3 * NPTS;
    float x0 = xb[n], x1 = xb[NPTS + n], x2 = xb[2 * NPTS + n];

    double acc[18];
    acc[0] = acc[1] = acc[2] = acc[3] = acc[4] = acc[5] = acc[6] = acc[7] = acc[8] = 0.0;
    acc[9]  = x0;       acc[10] = x1;       acc[11] = x2;
    acc[12] = (double)x0 * x0; acc[13] = (double)x0 * x1; acc[14] = (double)x0 * x2;
    acc[15] = (double)x1 * x1; acc[16] = (double)x1 * x2; acc[17] = (double)x2 * x2;

    const int* ip = idx + (size_t)col * KNN;
    for (int k = 0; k < KNN; ++k) {
        int j = ip[k];
        float f0 = xb[j] - x0, f1 = xb[NPTS + j] - x1, f2 = xb[2 * NPTS + j] - x2;
        acc[0] += f0; acc[1] += f1; acc[2] += f2;
        acc[3] += (double)f0 * f0; acc[4] += (double)f0 * f1; acc[5] += (double)f0 * f2;
        acc[6] += (double)f1 * f1; acc[7] += (double)f1 * f2; acc[8] += (double)f2 * f2;
    }

    __shared__ double sh[256];
    for (int v = 0; v < 18; ++v) {
        sh[tid] = acc[v];
        __syncthreads();
        for (int s = 128; s > 0; s >>= 1) {
            if (tid < s) sh[tid] += sh[tid + s];
            __syncthreads();
        }
        if (tid == 0) part[(size_t)blockIdx.x * 18 + v] = sh[0];
        __syncthreads();
    }
}

// ---------------------------------------------------------------------------
// Finalize input stats -> per-channel BN affine for bn (edge) and bn2 (self).
// aff layout: [0:64)=s1 [64:128)=t1 [128:192)=s2 [192:256)=t2.  1 block x 64.
// ---------------------------------------------------------------------------
__global__ void k_fin1(const double* __restrict__ part,
                       const float* __restrict__ Wnb, const float* __restrict__ Wself,
                       const float* __restrict__ bng, const float* __restrict__ bnb,
                       const float* __restrict__ bn2g, const float* __restrict__ bn2b,
                       float* __restrict__ aff) {
    __shared__ double tot[18];
    int tid = threadIdx.x;
    if (tid < 18) {
        double s = 0.0;
        for (int i = 0; i < 256; ++i) s += part[(size_t)i * 18 + tid];
        tot[tid] = s;
    }
    __syncthreads();
    int o = tid;
    {   // edge branch: BN over B*N*K
        double M = (double)COLS * (double)KNN;
        double w0 = Wnb[o * 3 + 0], w1 = Wnb[o * 3 + 1], w2 = Wnb[o * 3 + 2];
        double mean = (w0 * tot[0] + w1 * tot[1] + w2 * tot[2]) / M;
        double ey2 = (w0 * w0 * tot[3] + w1 * w1 * tot[6] + w2 * w2 * tot[8] +
                      2.0 * (w0 * w1 * tot[4] + w0 * w2 * tot[5] + w1 * w2 * tot[7])) / M;
        double var = ey2 - mean * mean;
        float s = bng[o] * rsqrtf((float)var + BN_EPS);
        aff[o] = s;
        aff[64 + o] = bnb[o] - (float)mean * s;
    }
    {   // self branch: BN over B*N
        double M = (double)COLS;
        double w0 = Wself[o * 3 + 0], w1 = Wself[o * 3 + 1], w2 = Wself[o * 3 + 2];
        double mean = (w0 * tot[9] + w1 * tot[10] + w2 * tot[11]) / M;
        double ey2 = (w0 * w0 * tot[12] + w1 * w1 * tot[15] + w2 * w2 * tot[17] +
                      2.0 * (w0 * w1 * tot[13] + w0 * w2 * tot[14] + w1 * w2 * tot[16])) / M;
        double var = ey2 - mean * mean;
        float s = bn2g[o] * rsqrtf((float)var + BN_EPS);
        aff[128 + o] = s;
        aff[192 + o] = bn2b[o] - (float)mean * s;
    }
}

// ---------------------------------------------------------------------------
// Pass 2: gather neighbors, 64x3 edge conv, BN+lrelu+max_k (via monotone
// commutation: max if scale>=0 else min), plus the self branch a2.
// block = 256 threads = 4 columns x 64 channels.
// a1/a2 stored in output layout (B,64,N).
// ---------------------------------------------------------------------------
__global__ void k_edge(const float* __restrict__ x, const int* __restrict__ idx,
                       const float* __restrict__ Wnb, const float* __restrict__ Wself,
                       const float* __restrict__ aff,
                       float* __restrict__ a1, float* __restrict__ a2) {
    __shared__ float nb[4][KNN][3];
    __shared__ float xc[4][3];
    int tid = threadIdx.x, sub = tid >> 6, o = tid & 63;
    int col = blockIdx.x * 4 + sub;
    int b = col >> 12, n = col & 4095;
    const float* xb = x + (size_t)b * 3 * NPTS;

    if (o < KNN) {
        int j = idx[(size_t)col * KNN + o];
        nb[sub][o][0] = xb[j];
        nb[sub][o][1] = xb[NPTS + j];
        nb[sub][o][2] = xb[2 * NPTS + j];
    } else if (o >= 60 && o < 63) {
        xc[sub][o - 60] = xb[(o - 60) * NPTS + n];
    }
    __syncthreads();

    float x0 = xc[sub][0], x1 = xc[sub][1], x2 = xc[sub][2];
    float w0 = Wnb[o * 3 + 0], w1 = Wnb[o * 3 + 1], w2 = Wnb[o * 3 + 2];
    float mx = -3.4e38f, mn = 3.4e38f;
    for (int k = 0; k < KNN; ++k) {
        float y = w0 * (nb[sub][k][0] - x0) + w1 * (nb[sub][k][1] - x1) +
                  w2 * (nb[sub][k][2] - x2);
        mx = fmaxf(mx, y);
        mn = fminf(mn, y);
    }
    float s1 = aff[o], t1 = aff[64 + o];
    float v = s1 * ((s1 >= 0.f) ? mx : mn) + t1;
    v = (v >= 0.f) ? v : 0.2f * v;
    size_t oidx = (((size_t)(b * 64 + o)) << 12) + n;
    a1[oidx] = v;

    float y2 = Wself[o * 3 + 0] * x0 + Wself[o * 3 + 1] * x1 + Wself[o * 3 + 2] * x2;
    float s2 = aff[128 + o], t2 = aff[192 + o];
    float v2 = s2 * y2 + t2;
    v2 = (v2 >= 0.f) ? v2 : 0.2f * v2;
    a2[oidx] = v2;
}

// ---------------------------------------------------------------------------
// 64x64 GEMM over 65536 columns via V_WMMA_F32_16X16X4_F32.
// One wave per 16x16 tile, K=64 in 16 steps of 4. U is in (B,64,N) layout so
// B-operand loads and C stores are lane-coalesced. ADD_TWO selects U0+U1 at
// compile time (no per-step branches around the WMMAs).
// grid = 2048 blocks x 256 threads (8 waves/block, 16384 tiles = 4 x 4096).
// ---------------------------------------------------------------------------
template <int ADD_TWO>
__global__ void k_gemm64(const float* __restrict__ A, const float* __restrict__ bias,
                         const float* __restrict__ U0, const float* __restrict__ U1,
                         float* __restrict__ Z) {
    int lane = threadIdx.x & 31;
    int wave = threadIdx.x >> 5;
    int tile = blockIdx.x * 8 + wave;
    int mt = tile & 3, nt = tile >> 2;
    int Mbase = mt << 4;
    int b = nt >> 8;                 // 4096/16 = 256 N-tiles per batch image
    int n0 = (nt & 255) << 4;
    int half = lane >> 4, l = lane & 15;

    v8f c = {0.f, 0.f, 0.f, 0.f, 0.f, 0.f, 0.f, 0.f};
    const float* Arow = A + (size_t)(Mbase + l) * 64 + 2 * half;
    size_t colbase = ((size_t)(b * 64)) << 12;   // b*64*4096 floats
    int nloc = n0 + l;

#pragma unroll
    for (int kk = 0; kk < 64; kk += 4) {
        // A fragment: lane<16 holds K = kk,kk+1 ; lane>=16 holds K = kk+2,kk+3
        v2f af = *(const v2f*)(Arow + kk);
        // B fragment: rows (kk + 2*half, +1) at column (Nbase + l)
        int r0 = kk + 2 * half;
        size_t ad0 = colbase + (((size_t)r0) << 12) + nloc;
        float b0 = U0[ad0], b1 = U0[ad0 + 4096];
        if (ADD_TWO) { b0 += U1[ad0]; b1 += U1[ad0 + 4096]; }
        v2f bf = {b0, b1};
        c = __builtin_amdgcn_wmma_f32_16x16x4_f32(false, af, false, bf,
                                                  (short)0, c, false, false);
    }
#pragma unroll
    for (int v = 0; v < 8; ++v) {
        int row = Mbase + v + 8 * half;
        Z[colbase + (((size_t)row) << 12) + nloc] = c[v] + bias[row];
    }
}

// ---------------------------------------------------------------------------
// Per-channel sum / sumsq over the 65536 columns of a (B,64,N) activation.
// grid = 512 blocks (64 channels x 8 chunks) x 256 threads; fp64, tree reduce.
// ---------------------------------------------------------------------------
__global__ void k_colstats(const float* __restrict__ Z, double* __restrict__ part) {
    int r = blockIdx.x & 63, chunk = blockIdx.x >> 6;
    int tid = threadIdx.x;
    double s = 0.0, q = 0.0;
    int c0 = chunk * 8192, c1 = c0 + 8192;
    for (int col = c0 + tid; col < c1; col += 256) {
        int b = col >> 12, n = col & 4095;
        float v = Z[(((size_t)(b * 64 + r)) << 12) + n];
        s += v;
        q += (double)v * v;
    }
    __shared__ double sh[256];
    sh[tid] = s;
    __syncthreads();
    for (int st = 128; st > 0; st >>= 1) {
        if (tid < st) sh[tid] += sh[tid + st];
        __syncthreads();
    }
    if (tid == 0) part[(size_t)blockIdx.x * 2] = sh[0];
    __syncthreads();
    sh[tid] = q;
    __syncthreads();
    for (int st = 128; st > 0; st >>= 1) {
        if (tid < st) sh[tid] += sh[tid + st];
        __syncthreads();
    }
    if (tid == 0) part[(size_t)blockIdx.x * 2 + 1] = sh[0];
}

// Finalize channel stats -> BN affine (s in aff[0:64), t in aff[64:128)).
__global__ void k_finbn(const double* __restrict__ part,
                        const float* __restrict__ g, const float* __restrict__ bb,
                        float* __restrict__ aff) {
    int o = threadIdx.x;
    double S = 0.0, Q = 0.0;
    for (int c = 0; c < 8; ++c) {
        S += part[(size_t)(c * 64 + o) * 2];
        Q += part[(size_t)(c * 64 + o) * 2 + 1];
    }
    double mean = S / (double)COLS;
    double var = Q / (double)COLS - mean * mean;
    float s = g[o] * rsqrtf((float)var + BN_EPS);
    aff[o] = s;
    aff[64 + o] = bb[o] - (float)mean * s;
}

// Fold fc's BN into fco's weights: W2 = fco_w * diag(s), b2 = fco_w @ t + fco_b.
__global__ void k_fold(const float* __restrict__ fcoW, const float* __restrict__ fcoB,
                       const float* __restrict__ affZ,
                       float* __restrict__ W2, float* __restrict__ b2) {
    int o = threadIdx.x;
    float acc = fcoB[o];
    for (int c = 0; c < 64; ++c) {
        float w = fcoW[o * 64 + c];
        W2[o * 64 + c] = w * affZ[c];
        acc += w * affZ[64 + c];
    }
    b2[o] = acc;
}

// Final fused sigmoid gate: out = a2 + sigmoid(BN(cat_pre)) * (a1 - a2).
// All buffers in identical (B,64,N) layout -> fully coalesced.
__global__ void k_final(const float* __restrict__ a1, const float* __restrict__ a2,
                        const float* __restrict__ cp, const float* __restrict__ affC,
                        float* __restrict__ out) {
    size_t i = (size_t)blockIdx.x * 256 + threadIdx.x;
    int o = (int)((i >> 12) & 63);
    float cat = cp[i] * affC[o] + affC[64 + o];
    float att = 1.f / (1.f + expf(-cat));
    float v1 = a1[i], v2 = a2[i];
    out[i] = v2 + att * (v1 - v2);
}

// ---------------------------------------------------------------------------
extern "C" void kernel_launch(void* const* d_in, const int* in_sizes, int n_in,
                              void* d_out, int out_size, void* d_ws, size_t ws_size,
                              hipStream_t stream) {
    const float* x      = (const float*)d_in[0];
    const int*   idx    = (const int*)d_in[1];
    const float* Wnb    = (const float*)d_in[2];
    const float* Wself  = (const float*)d_in[3];
    const float* bng    = (const float*)d_in[4];
    const float* bnb    = (const float*)d_in[5];
    const float* bn2g   = (const float*)d_in[6];
    const float* bn2b   = (const float*)d_in[7];
    const float* fcw    = (const float*)d_in[8];
    const float* fcb    = (const float*)d_in[9];
    const float* fcbng  = (const float*)d_in[10];
    const float* fcbnb  = (const float*)d_in[11];
    const float* fcow   = (const float*)d_in[12];
    const float* fcob   = (const float*)d_in[13];
    const float* fcobng = (const float*)d_in[14];
    const float* fcobnb = (const float*)d_in[15];

    char* ws = (char*)d_ws;
    const size_t ABUF = (size_t)COLS * CH * sizeof(float);   // 16 MB
    float* a1 = (float*)ws;
    float* a2 = (float*)(ws + ABUF);
    float* zp = (float*)(ws + 2 * ABUF);
    float* cp = (float*)(ws + 3 * ABUF);
    char* p = ws + 4 * ABUF;
    double* P1    = (double*)p; p += 256 * 18 * sizeof(double);
    float*  aff1  = (float*)p;  p += 256 * sizeof(float);
    double* partZ = (double*)p; p += 512 * 2 * sizeof(double);
    float*  affZ  = (float*)p;  p += 128 * sizeof(float);
    float*  W2    = (float*)p;  p += 4096 * sizeof(float);
    float*  b2    = (float*)p;  p += 64 * sizeof(float);
    double* partC = (double*)p; p += 512 * 2 * sizeof(double);
    float*  affC  = (float*)p;

    k_stats_in<<<256, 256, 0, stream>>>(x, idx, P1);
    k_fin1<<<1, 64, 0, stream>>>(P1, Wnb, Wself, bng, bnb, bn2g, bn2b, aff1);
    k_edge<<<COLS / 4, 256, 0, stream>>>(x, idx, Wnb, Wself, aff1, a1, a2);
    k_gemm64<1><<<2048, 256, 0, stream>>>(fcw, fcb, a1, a2, zp);
    k_colstats<<<512, 256, 0, stream>>>(zp, partZ);
    k_finbn<<<1, 64, 0, stream>>>(partZ, fcbng, fcbnb, affZ);
    k_fold<<<1, 64, 0, stream>>>(fcow, fcob, affZ, W2, b2);
    k_gemm64<0><<<2048, 256, 0, stream>>>(W2, b2, zp, zp, cp);
    k_colstats<<<512, 256, 0, stream>>>(cp, partC);
    k_finbn<<<1, 64, 0, stream>>>(partC, fcobng, fcobnb, affC);
    k_final<<<(size_t)COLS * CH / 256, 256, 0, stream>>>(a1, a2, cp, affC, (float*)d_out);
}